// GCN_9818295239343
// MI455X (gfx1250) — compile-verified
//
#include <hip/hip_runtime.h>
#include <hip/hip_bf16.h>
#include <math.h>

typedef __attribute__((ext_vector_type(2))) float v2f;
typedef __attribute__((ext_vector_type(8))) float v8f;

#define F_IN  128
#define F_HID 64
#define F_OUT 40

// ---------------------------------------------------------------------------
// Zero-init (vectorized): agg1 (N*64) and out (N*40) accumulators
// counts are in float4 units
// ---------------------------------------------------------------------------
__global__ void zero2_kernel(float4* __restrict__ a, int na4,
                             float4* __restrict__ b, int nb4) {
    int idx = blockIdx.x * blockDim.x + threadIdx.x;
    float4 z = make_float4(0.f, 0.f, 0.f, 0.f);
    if (idx < na4) a[idx] = z;
    if (idx < nb4) b[idx] = z;
}

// ---------------------------------------------------------------------------
// GEMM1: xw[N,64] = x[N,128] @ W1[128,64]   (fp32 WMMA 16x16x4)
// grid = N/16 blocks, 128 threads (4 waves); wave w computes cols [16w,16w+16)
// ---------------------------------------------------------------------------
__global__ void gemm1_wmma(const float* __restrict__ x,
                           const float* __restrict__ W1,
                           float* __restrict__ xw) {
    const int rowTile = blockIdx.x;
    const int wave = threadIdx.x >> 5;
    const int lane = threadIdx.x & 31;
    const int half = lane >> 4;        // 0: K pair {0,1}, 1: K pair {2,3}
    const int l16  = lane & 15;
    const int row0 = rowTile * 16;
    const int n0   = wave * 16;

    const float* arow = x  + (size_t)(row0 + l16) * F_IN + 2 * half;   // A: row l16
    const float* bcol = W1 + (size_t)(2 * half) * F_HID + (n0 + l16);  // B: col l16

    v8f c = {};
    for (int k = 0; k < F_IN; k += 4) {
        v2f a, b;
        a.x = arow[k];
        a.y = arow[k + 1];
        b.x = bcol[(size_t)k * F_HID];
        b.y = bcol[(size_t)(k + 1) * F_HID];
        c = __builtin_amdgcn_wmma_f32_16x16x4_f32(false, a, false, b,
                                                  (short)0, c, false, false);
    }
    float* op = xw + (size_t)(row0 + 8 * half) * F_HID + n0 + l16;
#pragma unroll
    for (int v = 0; v < 8; ++v) op[(size_t)v * F_HID] = c[v];
}

// ---------------------------------------------------------------------------
// Scatter (64 feats): agg[dst[e]][f] += feat[src[e]][f]
// one thread per (edge, 4-feature chunk): b128 loads, 16 threads/edge
// ---------------------------------------------------------------------------
__global__ void scatter64_kernel(const float4* __restrict__ feat4,
                                 const int* __restrict__ ei, int E,
                                 float* __restrict__ agg) {
    unsigned idx = blockIdx.x * blockDim.x + threadIdx.x;
    unsigned total = (unsigned)E * 16u;
    if (idx >= total) return;
    int e = idx >> 4;
    int q = idx & 15;            // 4-feature chunk
    int s = ei[e];               // src row
    int d = ei[E + e];           // dst row
    float4 v = feat4[(size_t)s * 16 + q];
    float* ap = agg + (size_t)d * F_HID + q * 4;
    atomicAdd(ap + 0, v.x);
    atomicAdd(ap + 1, v.y);
    atomicAdd(ap + 2, v.z);
    atomicAdd(ap + 3, v.w);
}

// ---------------------------------------------------------------------------
// h = relu(agg + b1), in-place, vectorized (N*16 float4 chunks)
// ---------------------------------------------------------------------------
__global__ void relu_bias_kernel(float4* __restrict__ h4,
                                 const float4* __restrict__ b14, int n4) {
    int idx = blockIdx.x * blockDim.x + threadIdx.x;
    if (idx >= n4) return;
    float4 v = h4[idx];
    float4 b = b14[idx & 15];    // 64 feats = 16 float4
    v.x = fmaxf(v.x + b.x, 0.f);
    v.y = fmaxf(v.y + b.y, 0.f);
    v.z = fmaxf(v.z + b.z, 0.f);
    v.w = fmaxf(v.w + b.w, 0.f);
    h4[idx] = v;
}

// ---------------------------------------------------------------------------
// GEMM2: h2[N,40] = h[N,64] @ W2[64,40]   (fp32 WMMA 16x16x4)
// grid = N/16 blocks, 96 threads (3 waves); store guard col<40
// ---------------------------------------------------------------------------
__global__ void gemm2_wmma(const float* __restrict__ h,
                           const float* __restrict__ W2,
                           float* __restrict__ h2) {
    const int rowTile = blockIdx.x;
    const int wave = threadIdx.x >> 5;
    const int lane = threadIdx.x & 31;
    const int half = lane >> 4;
    const int l16  = lane & 15;
    const int row0 = rowTile * 16;
    const int n0   = wave * 16;
    const int col  = n0 + l16;
    const int bcol = col < F_OUT ? col : (F_OUT - 1);  // clamp B reads; cols>=40 unstored

    const float* arow = h  + (size_t)(row0 + l16) * F_HID + 2 * half;
    const float* bptr = W2 + (size_t)(2 * half) * F_OUT + bcol;

    v8f c = {};
    for (int k = 0; k < F_HID; k += 4) {
        v2f a, b;
        a.x = arow[k];
        a.y = arow[k + 1];
        b.x = bptr[(size_t)k * F_OUT];
        b.y = bptr[(size_t)(k + 1) * F_OUT];
        c = __builtin_amdgcn_wmma_f32_16x16x4_f32(false, a, false, b,
                                                  (short)0, c, false, false);
    }
    if (col < F_OUT) {
        float* op = h2 + (size_t)(row0 + 8 * half) * F_OUT + col;
#pragma unroll
        for (int v = 0; v < 8; ++v) op[(size_t)v * F_OUT] = c[v];
    }
}

// ---------------------------------------------------------------------------
// Scatter (40 feats): out[dst[e]][f] += h2[src[e]][f]
// 10 float4 chunks per edge; 320 threads/block = 32 edges/block
// row base 40*r floats = 160*r bytes -> 16B aligned, b128 loads OK
// ---------------------------------------------------------------------------
__global__ void scatter40_kernel(const float* __restrict__ feat,
                                 const int* __restrict__ ei, int E,
                                 float* __restrict__ out) {
    int t = threadIdx.x;
    int e = blockIdx.x * 32 + t / 10;
    int q = t % 10;              // 4-feature chunk
    if (e >= E) return;
    int s = ei[e];
    int d = ei[E + e];
    float4 v = *(const float4*)(feat + (size_t)s * F_OUT + q * 4);
    float* ap = out + (size_t)d * F_OUT + q * 4;
    atomicAdd(ap + 0, v.x);
    atomicAdd(ap + 1, v.y);
    atomicAdd(ap + 2, v.z);
    atomicAdd(ap + 3, v.w);
}

// ---------------------------------------------------------------------------
// out = log_softmax(out + b2) per row of 40.  One wave (32 lanes) per row:
// lane handles f=lane and (lane<8) f=lane+32.
// ---------------------------------------------------------------------------
__global__ void logsoftmax_kernel(float* __restrict__ out,
                                  const float* __restrict__ b2, int N) {
    int row  = blockIdx.x * 8 + (threadIdx.x >> 5);
    int lane = threadIdx.x & 31;
    if (row >= N) return;

    float* r = out + (size_t)row * F_OUT;
    float v1 = r[lane] + b2[lane];
    float v2 = (lane < 8) ? (r[32 + lane] + b2[32 + lane]) : -INFINITY;

    float m = fmaxf(v1, v2);
#pragma unroll
    for (int off = 16; off > 0; off >>= 1)
        m = fmaxf(m, __shfl_xor(m, off, 32));

    float s = expf(v1 - m) + ((lane < 8) ? expf(v2 - m) : 0.0f);
#pragma unroll
    for (int off = 16; off > 0; off >>= 1)
        s += __shfl_xor(s, off, 32);

    float ls = logf(s);
    r[lane] = v1 - m - ls;
    if (lane < 8) r[32 + lane] = v2 - m - ls;
}

// ---------------------------------------------------------------------------
extern "C" void kernel_launch(void* const* d_in, const int* in_sizes, int n_in,
                              void* d_out, int out_size, void* d_ws, size_t ws_size,
                              hipStream_t stream) {
    const float* x  = (const float*)d_in[0];
    const int*   ei = (const int*)d_in[1];
    const float* W1 = (const float*)d_in[2];
    const float* b1 = (const float*)d_in[3];
    const float* W2 = (const float*)d_in[4];
    const float* b2 = (const float*)d_in[5];
    float* out = (float*)d_out;

    const int N = in_sizes[0] / F_IN;     // 100000
    const int E = in_sizes[1] / 2;        // 1,600,000

    float* xw1  = (float*)d_ws;           // N*64 floats
    float* agg1 = xw1 + (size_t)N * F_HID;
    float* h2   = agg1 + (size_t)N * F_HID;

    const int rowTiles = N / 16;          // 6250 (N divisible by 16)

    // zero accumulators (agg1 and out), float4-wide
    {
        int na4 = N * F_HID / 4, nb4 = N * F_OUT / 4;
        int n = na4 > nb4 ? na4 : nb4;
        zero2_kernel<<<(n + 255) / 256, 256, 0, stream>>>(
            (float4*)agg1, na4, (float4*)out, nb4);
    }

    // layer 1
    gemm1_wmma<<<rowTiles, 128, 0, stream>>>(x, W1, xw1);
    {
        unsigned total = (unsigned)E * 16u;   // 4-feature chunks
        scatter64_kernel<<<(total + 255) / 256, 256, 0, stream>>>(
            (const float4*)xw1, ei, E, agg1);
    }
    relu_bias_kernel<<<(N * 16 + 255) / 256, 256, 0, stream>>>(
        (float4*)agg1, (const float4*)b1, N * 16);

    // layer 2
    gemm2_wmma<<<rowTiles, 96, 0, stream>>>(agg1, W2, h2);
    scatter40_kernel<<<(E + 31) / 32, 320, 0, stream>>>(h2, ei, E, out);

    // bias + log_softmax (in place on out)
    logsoftmax_kernel<<<(N + 7) / 8, 256, 0, stream>>>(out, b2, N);
}